// ChamfersDistance_16741782520028
// MI455X (gfx1250) — compile-verified
//
#include <hip/hip_runtime.h>
#include <math.h>

// Pairwise-L2 Chamfer (buggy-reference variant: both terms use min over dim 0).
// dist^2 tiles via the only f32 WMMA shape on CDNA5, V_WMMA_F32_16X16X4_F32,
// using the norm-augmentation trick:
//   A'[i]   = (a0, a1, a2, ||a||^2)        (16x4, K padded from 3 to 4)
//   B'[:,j] = (-2b0, -2b1, -2b2, 1)^T      (4x16)
//   C init  = ||b_j||^2 broadcast per lane (N = lane%16 in the C layout)
//   D = A' x B' + C = ||a_i||^2 + ||b_j||^2 - 2 a.b = dist^2   (16x16 tile)

typedef __attribute__((ext_vector_type(2))) float v2f;
typedef __attribute__((ext_vector_type(8))) float v8f;

#define N_PTS 8192
#define KDIM 3
#define NT (N_PTS / 16)               // 512 tiles along each axis
#define ICHUNKS 8                     // i-range split for occupancy
#define ITILES_PER_CHUNK (NT / ICHUNKS)

__global__ void cham_init_ws(unsigned int* __restrict__ ws) {
    int t = blockIdx.x * blockDim.x + threadIdx.x;
    if (t < N_PTS) ws[t] = 0x7F800000u;  // +inf bit pattern
}

__device__ __forceinline__ float tile_min8(v8f d) {
    // Tree-min of the 8 C/D VGPRs; fuses into v_min3_num_f32 chains and only
    // the fresh WMMA outputs need canonicalization (fmin results are canonical).
    float t0 = fminf(d[0], d[1]);
    float t1 = fminf(d[2], d[3]);
    float t2 = fminf(d[4], d[5]);
    float t3 = fminf(d[6], d[7]);
    return fminf(fminf(t0, t1), fminf(t2, t3));
}

__global__ void cham_min_kernel(const float* __restrict__ A,
                                const float* __restrict__ B,
                                unsigned int* __restrict__ ws) {
    const int wave = (blockIdx.x * blockDim.x + threadIdx.x) >> 5;
    const int lane = threadIdx.x & 31;
    const int half = lane >> 4;   // which 16-lane half of the wave
    const int ln   = lane & 15;

    const int jt    = wave & (NT - 1);   // j-tile (columns of the distance matrix)
    const int chunk = wave >> 9;         // i-chunk: [chunk*64, chunk*64+64) i-tiles
    const int j     = jt * 16 + ln;

    // ---- Build B' (4x16) once per wave; fold the -2 into B. ----
    // B layout (mirrors the ISA's 32-bit A 16x4 pattern): N = lane%16;
    // lanes 0-15 hold rows K=0 (v0), K=1 (v1); lanes 16-31 hold K=2 (v0), K=3 (v1).
    const float b0 = B[j * KDIM + 0];
    const float b1 = B[j * KDIM + 1];
    const float b2 = B[j * KDIM + 2];
    const float nb = b0 * b0 + b1 * b1 + b2 * b2;

    v2f bm;
    bm.x = half ? (-2.0f * b2) : (-2.0f * b0);
    bm.y = half ? 1.0f         : (-2.0f * b1);

    v8f c;
#pragma unroll
    for (int k = 0; k < 8; ++k) c[k] = nb;

    // Single scalar running min per lane (min over this half's 8 M-rows,
    // all tiles in this wave's i-chunk, for column N = lane%16).
    float macc = 3.402823466e38f;  // FLT_MAX

    const int it0 = chunk * ITILES_PER_CHUNK;
#pragma unroll 2
    for (int it = it0; it < it0 + ITILES_PER_CHUNK; ++it) {
        const int i = it * 16 + ln;
        const float a0 = A[i * KDIM + 0];
        const float a1 = A[i * KDIM + 1];
        const float a2 = A[i * KDIM + 2];
        const float na = a0 * a0 + a1 * a1 + a2 * a2;

        // A 16x4 layout: M = lane%16; lanes 0-15: v0=K0, v1=K1;
        // lanes 16-31: v0=K2, v1=K3 (= the norm-augmentation slot).
        v2f am;
        am.x = half ? a2 : a0;
        am.y = half ? na : a1;

        // 8 args: (neg_a, A, neg_b, B, c_mod, C, reuse_a, reuse_b)
        v8f d = __builtin_amdgcn_wmma_f32_16x16x4_f32(
            false, am, false, bm, (short)0, c, false, false);

        macc = fminf(macc, tile_min8(d));
    }

    // Merge lane halves (lane L and L+16 hold the same column N = L%16).
    // ds_swizzle SWAPX16: group-of-32 mode, xor_mask=0x10, and_mask=0x1F.
    const int other = __builtin_amdgcn_ds_swizzle(__float_as_int(macc), 0x401F);
    float m = fminf(macc, __int_as_float(other));
    m = fmaxf(m, 0.0f);  // clamp cancellation noise in na+nb-2dot

    // Order-independent (hence deterministic) merge across i-chunks:
    // bit-pattern min is monotone for non-negative floats.
    if (lane < 16) {
        atomicMin(&ws[j], __float_as_uint(m));
    }
}

__global__ void cham_reduce_kernel(const unsigned int* __restrict__ ws,
                                   float* __restrict__ out) {
    __shared__ float sdata[256];
    const int tid = threadIdx.x;
    float s = 0.0f;
    for (int jj = tid; jj < N_PTS; jj += 256) {
        s += sqrtf(__uint_as_float(ws[jj]));
    }
    sdata[tid] = s;
    __syncthreads();
    for (int off = 128; off > 0; off >>= 1) {
        if (tid < off) sdata[tid] += sdata[tid + off];
        __syncthreads();
    }
    // loss = mean(dist0) + mean(dist1) = 2 * sum / N
    if (tid == 0) out[0] = sdata[0] * (2.0f / (float)N_PTS);
}

extern "C" void kernel_launch(void* const* d_in, const int* in_sizes, int n_in,
                              void* d_out, int out_size, void* d_ws, size_t ws_size,
                              hipStream_t stream) {
    const float* A = (const float*)d_in[0];   // input1: (8192, 3) f32
    const float* B = (const float*)d_in[1];   // input2: (8192, 3) f32
    float* out = (float*)d_out;               // scalar f32
    unsigned int* ws = (unsigned int*)d_ws;   // 8192 u32 per-column min(dist^2) bits

    cham_init_ws<<<(N_PTS + 255) / 256, 256, 0, stream>>>(ws);

    // 512 j-tiles * 8 i-chunks = 4096 waves = 512 blocks of 256 threads.
    const int total_waves = NT * ICHUNKS;
    cham_min_kernel<<<(total_waves * 32) / 256, 256, 0, stream>>>(A, B, ws);

    cham_reduce_kernel<<<1, 256, 0, stream>>>(ws, out);
}